// EctPointsLayer_86784109183420
// MI455X (gfx1250) — compile-verified
//
#include <hip/hip_runtime.h>

typedef float v2f __attribute__((ext_vector_type(2)));
typedef float v8f __attribute__((ext_vector_type(8)));

#define S_STEPS 32
#define T_DIRS  32
#define WAVES   16
#define BLOCK   (WAVES * 32)

__device__ __forceinline__ int lower_bound_i(const int* __restrict__ a, int n, int key) {
    int lo = 0, hi = n;
    while (lo < hi) {
        int mid = (lo + hi) >> 1;
        if (a[mid] < key) lo = mid + 1; else hi = mid;
    }
    return lo;
}

__global__ __launch_bounds__(BLOCK)
void ect_points_kernel(const float* __restrict__ x,
                       const float* __restrict__ v,
                       const float* __restrict__ lin,
                       const int*   __restrict__ batch,
                       float*       __restrict__ out,
                       int N)
{
    __shared__ float lds[WAVES * S_STEPS * T_DIRS];   // 64 KB: per-wave partial tiles

    const int b    = blockIdx.x;          // segment id
    const int tid  = threadIdx.x;
    const int wave = tid >> 5;
    const int lane = tid & 31;
    const int tl   = lane & 15;           // t within a 16-wide half
    const int hi   = lane >> 4;           // 0 -> points j..j, 1 -> points j+8

    // Segment range in the sorted batch array (uniform per block).
    const int segStart = lower_bound_i(batch, N, b);
    const int segEnd   = lower_bound_i(batch, N, b + 1);

    // sigmoid(200*(lin_s - nh)) = rcp(1 + exp2(C*(nh - lin_s))), C = 200*log2(e)
    const float C = 200.0f * 1.4426950408889634f;
    float cls[S_STEPS];                   // C * lin[s]; uniform -> SGPRs
    #pragma unroll
    for (int s = 0; s < S_STEPS; ++s) cls[s] = C * lin[s];

    // B matrices (4x16 f32) for the two t-halves, 16x16x4 layout:
    //   VGPR0: lanes0-15 = row K=0, lanes16-31 = row K=2 (zero pad)
    //   VGPR1: lanes0-15 = row K=1, lanes16-31 = row K=3 (zero pad)
    v2f B0, B1;
    B0.x = (lane < 16) ? v[0 * T_DIRS + lane]      : 0.0f;   // sin(theta_t)
    B0.y = (lane < 16) ? v[1 * T_DIRS + lane]      : 0.0f;   // cos(theta_t)
    B1.x = (lane < 16) ? v[0 * T_DIRS + 16 + lane] : 0.0f;
    B1.y = (lane < 16) ? v[1 * T_DIRS + 16 + lane] : 0.0f;

    float acc0[S_STEPS], acc1[S_STEPS];
    #pragma unroll
    for (int s = 0; s < S_STEPS; ++s) { acc0[s] = 0.0f; acc1[s] = 0.0f; }

    // Each wave consumes 16-point tiles of this segment, round-robin.
    for (int tile0 = segStart + wave * 16; tile0 < segEnd; tile0 += WAVES * 16) {
        // Prefetch next tile this wave will touch (gfx1250 global_prefetch_b8).
        if (tile0 + WAVES * 16 < segEnd)
            __builtin_prefetch(&x[2 * (tile0 + WAVES * 16)], 0, 0);

        // A matrix (16x4 f32): lanes0-15 hold K=0,1 = (x0,x1) of point tile0+lane;
        // lanes16-31 hold K=2,3 = zero padding.
        int pA  = tile0 + lane;
        int pAc = (pA < segEnd) ? pA : (segEnd - 1);   // clamp address; masked below
        v2f A;
        if (lane < 16) {
            A.x = x[2 * pAc + 0];
            A.y = x[2 * pAc + 1];
        } else {
            A.x = 0.0f;
            A.y = 0.0f;
        }

        v8f Cz = {};
        // nh tiles: D[j] (lane) = <x_p, v_t>, p = tile0 + j + 8*(lane>=16), t = lane%16 (+16)
        v8f D0 = __builtin_amdgcn_wmma_f32_16x16x4_f32(false, A, false, B0,
                                                       (short)0, Cz, false, false);
        v8f D1 = __builtin_amdgcn_wmma_f32_16x16x4_f32(false, A, false, B1,
                                                       (short)0, Cz, false, false);

        #pragma unroll
        for (int j = 0; j < 8; ++j) {
            int p = tile0 + j + (hi << 3);
            if (p < segEnd) {
                float base0 = C * D0[j];   // C * nh  (t half 0)
                float base1 = C * D1[j];   // C * nh  (t half 1)
                #pragma unroll
                for (int s = 0; s < S_STEPS; ++s) {
                    float a0 = base0 - cls[s];
                    float a1 = base1 - cls[s];
                    acc0[s] += __builtin_amdgcn_rcpf(1.0f + __builtin_amdgcn_exp2f(a0));
                    acc1[s] += __builtin_amdgcn_rcpf(1.0f + __builtin_amdgcn_exp2f(a1));
                }
            }
        }
    }

    // Fold the two lane-halves (same t, disjoint points): commutative 2-way add,
    // bitwise deterministic. Lanes l and l+16 end up with identical sums.
    float* slot = &lds[wave * (S_STEPS * T_DIRS)];
    #pragma unroll
    for (int s = 0; s < S_STEPS; ++s) {
        float t0 = acc0[s] + __shfl_xor(acc0[s], 16, 32);
        float t1 = acc1[s] + __shfl_xor(acc1[s], 16, 32);
        if (lane < 16) {
            slot[s * T_DIRS + tl]      = t0;
            slot[s * T_DIRS + 16 + tl] = t1;
        }
    }
    __syncthreads();

    // Deterministic fixed-order reduction of the 16 wave partials, then store.
    for (int i = tid; i < S_STEPS * T_DIRS; i += BLOCK) {
        float sum = 0.0f;
        #pragma unroll
        for (int w = 0; w < WAVES; ++w)
            sum += lds[w * (S_STEPS * T_DIRS) + i];
        out[b * (S_STEPS * T_DIRS) + i] = sum;
    }
}

extern "C" void kernel_launch(void* const* d_in, const int* in_sizes, int n_in,
                              void* d_out, int out_size, void* d_ws, size_t ws_size,
                              hipStream_t stream) {
    const float* x     = (const float*)d_in[0];   // [N,2]
    const float* v     = (const float*)d_in[1];   // [2,T]
    const float* lin   = (const float*)d_in[2];   // [S,1,1]
    const int*   batch = (const int*)d_in[3];     // [N], sorted
    (void)d_ws; (void)ws_size; (void)n_in;

    const int N = in_sizes[0] / 2;
    const int B = out_size / (S_STEPS * T_DIRS); // [B,S,T]

    ect_points_kernel<<<B, BLOCK, 0, stream>>>(x, v, lin, batch, (float*)d_out, N);
}